// CONV_NODE_67053029425756
// MI455X (gfx1250) — compile-verified
//
#include <hip/hip_runtime.h>
#include <math.h>

// TBCNN conv for MI455X (gfx1250, wave32).
// Shapes fixed by the reference: B=16, N=2048, C=16, F=128, O=128.
//
// out[b,n,o] = tanh( b_conv[o] + sum_{j=0..383} RES[b,n,j] * Wcat[j,o] )
//   RES[b,n, f*3+0] = nodes[b,n,f]
//   RES[b,n, f*3+1] = sum_c c_r[b,n,c] * lookup[b, children[b,n,c], f]
//   RES[b,n, f*3+2] = sum_c c_l[b,n,c] * lookup[b, children[b,n,c], f]
//   Wcat rows 0..127 = w_t, 128..255 = w_r, 256..383 = w_l
// (the reference's reshape (F,3)->(3,F) is a flat reinterpret, making the
//  contraction exactly this K=384 fp32 GEMM; fp32 WMMA keeps numerics.)

typedef __attribute__((ext_vector_type(2))) float v2f;
typedef __attribute__((ext_vector_type(8))) float v8f;

static constexpr int kB = 16;
static constexpr int kN = 2048;
static constexpr int kC = 16;
static constexpr int kF = 128;
static constexpr int kO = 128;
static constexpr int kMTile = 32;                  // nodes per block (2 WMMA M-tiles)
static constexpr int kTilesPerBatch = kN / kMTile; // 64
static constexpr int kResStride = 3 * kF + 4;      // 388 floats: bank-conflict pad
static constexpr int kOutStride = kO + 4;          // 132 floats: staging pad

__global__ __launch_bounds__(256) void tbcnn_conv_wmma(
    const float* __restrict__ nodes,      // [B,N,F]
    const int*   __restrict__ children,   // [B,N,C]
    const float* __restrict__ w_t,        // [F,O]
    const float* __restrict__ w_r,        // [F,O]
    const float* __restrict__ w_l,        // [F,O]
    const float* __restrict__ b_conv,     // [O]
    float*       __restrict__ out)        // [B,N,O]
{
  __shared__ int   s_ch[kMTile][kC];
  __shared__ float s_cr[kMTile][kC];
  __shared__ float s_cl[kMTile][kC];
  __shared__ float s_res[kMTile][kResStride];      // ~49.7 KB (also reused as out stage)

  const int tid  = threadIdx.x;                    // 0..255, 8 wave32
  const int b    = blockIdx.x >> 6;                // / kTilesPerBatch
  const int tile = blockIdx.x & (kTilesPerBatch - 1);
  const int node_base = tile * kMTile;
  const float* __restrict__ nb = nodes + (size_t)b * kN * kF;

  // ---------- Phase A: child indices + per-child coefficients ----------
  for (int e = tid; e < kMTile * kC; e += 256) {
    const int node = e >> 4;
    const int c    = e & 15;
    s_ch[node][c] = children[((size_t)b * kN + (node_base + node)) * kC + c];
  }
  __syncthreads();
  for (int e = tid; e < kMTile * kC; e += 256) {
    const int node = e >> 4;
    const int c    = e & 15;
    int ns = 0;
#pragma unroll
    for (int j = 0; j < kC; ++j) ns += (s_ch[node][j] != 0) ? 1 : 0;
    const float mask = (s_ch[node][c] != 0) ? 1.0f : 0.0f;
    float cr;
    if (ns == 1) {
      cr = (c == 0) ? 0.5f : 0.0f;                 // reference "singles" branch
    } else {
      const float denom = (float)ns - 1.0f;
      const float safe  = (denom == 0.0f) ? 1.0f : denom;
      cr = (float)c * mask / safe;
    }
    const float cl = (1.0f - cr) * mask;
    s_cr[node][c] = cr;
    s_cl[node][c] = cl;
  }
  __syncthreads();

  // ---------- Phase B: gather children, build RES tile in LDS ----------
  {
    const int f = tid & 127;                       // coalesced 512B per child row
    for (int node = tid >> 7; node < kMTile; node += 2) {
      const int n = node_base + node;
      const float t = nb[(size_t)n * kF + f];
      float r = 0.0f, l = 0.0f;
#pragma unroll
      for (int c = 0; c < kC; ++c) {
        const int ch = s_ch[node][c];
        if (ch != 0) {                             // lookup row 0 is zeroed
          const float v = nb[(size_t)ch * kF + f];
          r = fmaf(s_cr[node][c], v, r);
          l = fmaf(s_cl[node][c], v, l);
        }
      }
      s_res[node][f * 3 + 0] = t;
      s_res[node][f * 3 + 1] = r;
      s_res[node][f * 3 + 2] = l;
    }
  }
  __syncthreads();

  // ---------- Phase C: two 16x16 D tiles per wave, shared B fragments ----------
  const int wave = tid >> 5;        // 0..7 -> output-column tile
  const int lane = tid & 31;
  const int M    = lane & 15;       // A row
  const int hi   = lane >> 4;       // lane group: K pair select / D row half
  const int col  = wave * 16 + (lane & 15);

  v8f acc0 = {};                    // nodes [0,16)
  v8f acc1 = {};                    // nodes [16,32)
  const float* Ws[3] = { w_t, w_r, w_l };
#pragma unroll
  for (int s = 0; s < 3; ++s) {
    const float* __restrict__ W  = Ws[s] + col;    // column 'col' of 128-row slab
    const float* __restrict__ R0 = &s_res[M][s * kF];
    const float* __restrict__ R1 = &s_res[M + 16][s * kF];
#pragma unroll 4
    for (int kk = 0; kk < kF; kk += 4) {
      // B fragment: bb[j] = Wcat[s*128 + kk + j + 2*hi, col]  (shared by both tiles)
      v2f bb;
      bb.x = W[(size_t)(kk + 2 * hi + 0) * kO];
      bb.y = W[(size_t)(kk + 2 * hi + 1) * kO];
      // A fragments: a[j] = A[M, kk + j + 2*hi]   (ISA 32-bit A 16x4 layout)
      v2f a0, a1;
      a0.x = R0[kk + 2 * hi + 0];
      a0.y = R0[kk + 2 * hi + 1];
      a1.x = R1[kk + 2 * hi + 0];
      a1.y = R1[kk + 2 * hi + 1];
      acc0 = __builtin_amdgcn_wmma_f32_16x16x4_f32(
          false, a0, false, bb, (short)0, acc0, false, false);
      acc1 = __builtin_amdgcn_wmma_f32_16x16x4_f32(
          false, a1, false, bb, (short)0, acc1, false, false);
    }
  }

  // ---------- Epilogue: tanh(acc + bias), stage in LDS, coalesced writeback ----------
  const float bias = b_conv[col];
  __syncthreads();                                 // all A reads done before reuse
  float* s_out = &s_res[0][0];                     // [kMTile][kOutStride] view
#pragma unroll
  for (int v = 0; v < 8; ++v) {
    const int m = v + 8 * hi;                      // C/D layout: acc[v] = D[m, lane&15]
    s_out[(size_t)m * kOutStride + col]        = tanhf(acc0[v] + bias);
    s_out[(size_t)(m + 16) * kOutStride + col] = tanhf(acc1[v] + bias);
  }
  __syncthreads();
  float* ob = out + ((size_t)b * kN + node_base) * kO;
  for (int i = tid; i < kMTile * kO; i += 256) {   // 128B contiguous per wave-op
    const int node = i >> 7;
    const int o    = i & 127;
    ob[(size_t)node * kO + o] = s_out[(size_t)node * kOutStride + o];
  }
}

extern "C" void kernel_launch(void* const* d_in, const int* in_sizes, int n_in,
                              void* d_out, int out_size, void* d_ws, size_t ws_size,
                              hipStream_t stream) {
  (void)in_sizes; (void)n_in; (void)out_size; (void)d_ws; (void)ws_size;
  const float* nodes    = (const float*)d_in[0];
  const int*   children = (const int*)d_in[1];
  // d_in[2] = feature_size scalar (compile-time 128 here)
  const float* w_t      = (const float*)d_in[3];
  const float* w_r      = (const float*)d_in[4];
  const float* w_l      = (const float*)d_in[5];
  const float* b_conv   = (const float*)d_in[6];
  float* out = (float*)d_out;

  dim3 grid(kB * kTilesPerBatch);  // 1024 blocks, 32 nodes each
  dim3 block(256);                 // 8 wave32
  tbcnn_conv_wmma<<<grid, block, 0, stream>>>(nodes, children, w_t, w_r, w_l,
                                              b_conv, out);
}